// SIRSDESimulator_37074157699390
// MI455X (gfx1250) — compile-verified
//
#include <hip/hip_runtime.h>
#include <math.h>

// ---------------------------------------------------------------------------
// SIR SDE simulator for MI455X (gfx1250).
//
// Roofline: 164 MB of dW streamed once (7us @ 23.3 TB/s), ~0.6 GFLOP total,
// but only 256 waves of parallelism exist (time scan is sequential). So the
// win is decoupling data movement from the (latency-exposed) compute waves:
// the Tensor Data Mover DMAs each 50-step x 128-path dW tile (strided 2D,
// row stride 32KB) into LDS asynchronously, double-buffered, while waves
// integrate from LDS. Sync = s_wait_tensorcnt (wave 0) + workgroup barrier.
// ---------------------------------------------------------------------------

#define B_TOT   8192
#define STEPS   5000
#define TPB     128             // paths per block (4 waves)
#define CHUNK   50              // time steps per DMA tile
#define NCHUNK  (STEPS / CHUNK) // 100

typedef __attribute__((ext_vector_type(4))) unsigned int v4u;
typedef __attribute__((ext_vector_type(8))) int          v8i;
typedef __attribute__((ext_vector_type(4))) int          v4i;

// Issue one TDM 2D tile load: CHUNK rows x TPB floats, row stride B_TOT floats.
// D# layout per CDNA5 ISA ch.8 (groups 0/1; groups 2/3 + extra group zero for
// 2D tensors). This toolchain exposes the 6-arg builtin form.
__device__ __forceinline__ void tdm_load_tile(const float* gptr, unsigned lds_off) {
    unsigned long long ga = (unsigned long long)(size_t)gptr;
    v4u g0;
    g0.x = 1u;                                   // count=1 (valid), user mode, no gather
    g0.y = lds_off;                              // lds_addr (bytes)
    g0.z = (unsigned)(ga & 0xFFFFFFFFu);         // global_addr[31:0]
    g0.w = (unsigned)((ga >> 32) & 0x01FFFFFFu)  // global_addr[56:32]
         | 0x80000000u;                          // type=2 ("image") in bits[127:126]
    v8i g1;
    g1[0] = (2 << 16);                           // data_size=2 (4 bytes); mask=0; no pad/iter
    g1[1] = (int)((unsigned)B_TOT << 16);        // tensor_dim0[15:0] @ bits[63:48]
    g1[2] = (int)((unsigned)STEPS << 16);        // dim0 hi16 = 0 | tensor_dim1[15:0]
    g1[3] = (int)((unsigned)TPB   << 16);        // dim1 hi16 = 0 | tile_dim0 = 128
    g1[4] = CHUNK;                               // tile_dim1 = 50, tile_dim2 = 0
    g1[5] = B_TOT;                               // tensor_dim0_stride[31:0] (elements)
    g1[6] = 0;                                   // stride hi | tensor_dim1_stride lo
    g1[7] = 0;
    v4i z4 = {0, 0, 0, 0};
    v8i z8 = {0, 0, 0, 0, 0, 0, 0, 0};
    __builtin_amdgcn_tensor_load_to_lds(g0, g1, z4, z4, z8, 0);
}

__global__ __launch_bounds__(TPB)
void sir_sde_kernel(const float* __restrict__ cond,
                    const float* __restrict__ dW,
                    const float* __restrict__ u,
                    float* __restrict__ out)
{
    __shared__ float smem[2][CHUNK * TPB];      // 2 x 25.6 KB double buffer

    const int tid = threadIdx.x;
    const int p   = blockIdx.x * TPB + tid;

    // Per-path parameters (coalesced float4 load).
    const float4 cc = ((const float4*)cond)[p];
    const float inf_rate = cc.x, rec = cc.y, mr = cc.z, volp = cc.w;
    const float r0i = inf_rate / rec;

    float s = 0.99f, i = 0.01f, r = 0.0f, r0 = r0i;

    const float DTf     = 0.01f;
    const float sqrt_dt = sqrtf(DTf);           // matches jnp.sqrt(f32(0.01))
    const float dtmr    = DTf * mr;             // DT*mr (Python left-assoc)

    // Online stats over the 49 sampled points.
    float mx = -1.0f; int am = 0;               // x >= 1e-5, so first sample sets mx
    float Lprev = 0.0f;
    float mean = 0.0f, M2 = 0.0f; int nd = 0;   // Welford over 48 log-diffs

    const unsigned lds0 = (unsigned)(size_t)&smem[0][0];   // addrspace(3) offset
    const unsigned lds1 = (unsigned)(size_t)&smem[1][0];
    const float* gbase = dW + (size_t)blockIdx.x * TPB;

    // Prime the pipeline: chunk 0 -> buffer 0 (wave 0 only; TDM ignores EXEC,
    // so gate with control flow, not lane masking).
    if (tid < 32) {
        tdm_load_tile(gbase, lds0);
    }

    for (int c = 0; c < NCHUNK; ++c) {
        if (tid < 32) {
            if (c + 1 < NCHUNK) {
                // Prefetch next chunk into the other buffer, then wait until
                // only that newest DMA is outstanding (TDM ops complete in
                // order per wave) => chunk c has landed in LDS.
                tdm_load_tile(gbase + (size_t)(c + 1) * CHUNK * B_TOT,
                              (c & 1) ? lds0 : lds1);
                __builtin_amdgcn_s_wait_tensorcnt(1);
            } else {
                __builtin_amdgcn_s_wait_tensorcnt(0);
            }
        }
        __syncthreads();   // chunk c visible to all waves

        const float* buf = (c & 1) ? &smem[1][0] : &smem[0][0];

        #pragma unroll 10
        for (int st = 0; st < CHUNK; ++st) {
            const float dw = buf[st * TPB + tid];     // ds_load_b32, conflict-free
            const float ni = (r0 * rec) * s;          // newly infected
            const float nr = rec * i;                 // newly recovered
            const float s1  = s - DTf * ni;
            const float i1  = i + DTf * (ni - nr);
            const float r1  = r + DTf * nr;
            const float t1  = sqrtf(fabsf(r0)) * volp;
            const float r01 = (r0 + dtmr * (r0i - r0)) + t1 * (dw * sqrt_dt);
            s = s1; i = i1; r = r1; r0 = r01;
        }

        // Sample points: global step t = c*50+49; t % 100 == 99  <=>  c odd.
        // Included steps are 99..4899 (49 samples) => exclude c == 99.
        if ((c & 1) == 1 && c < (NCHUNK - 1)) {
            float xi = i;
            if (!isfinite(xi)) xi = 0.0f;             // nan_to_num
            xi = fmaxf(xi, 1e-5f);                    // clip(1e-5)
            const int k = (c - 1) >> 1;               // sample index 0..48
            if (xi > mx) { mx = xi; am = k; }         // first-occurrence argmax
            const float L = logf(xi);
            if (k > 0) {
                const float d = L - Lprev;
                nd += 1;
                const float delta = d - mean;
                mean += delta / (float)nd;
                M2   += delta * (d - mean);
            }
            Lprev = L;
        }
        __syncthreads();   // everyone done reading buf before it is re-filled
    }

    const float var    = M2 / 48.0f;                  // jnp.std ddof=0 over 48 diffs
    const float vol_o  = sqrtf(fmaxf(var, 0.0f));
    const float max_at = ((float)am + u[p]) / 49.0f;  // (argmax + u) / TIME_STEPS

    out[p * 3 + 0] = mx;
    out[p * 3 + 1] = max_at;
    out[p * 3 + 2] = vol_o;
}

extern "C" void kernel_launch(void* const* d_in, const int* in_sizes, int n_in,
                              void* d_out, int out_size, void* d_ws, size_t ws_size,
                              hipStream_t stream) {
    const float* cond = (const float*)d_in[0];   // (8192, 4)
    const float* dW   = (const float*)d_in[1];   // (5000, 8192)
    const float* u    = (const float*)d_in[2];   // (8192,)
    float* out        = (float*)d_out;           // (8192, 3)

    sir_sde_kernel<<<B_TOT / TPB, TPB, 0, stream>>>(cond, dW, u, out);
}